// InstanceDecoder_3040836846105
// MI455X (gfx1250) — compile-verified
//
#include <hip/hip_runtime.h>

typedef float v2f __attribute__((ext_vector_type(2)));
typedef float v8f __attribute__((ext_vector_type(8)));

#define G      16
#define NN     2048
#define DD     512
#define BT     128   // output block tile (128x128)
#define BK     32    // K step staged through LDS
#define LDSTR  34    // padded LDS row stride in floats (bank-conflict-free, 8B-aligned)

// --- Kernel 1: per-row squared norms -> d_ws ---------------------------------
__global__ __launch_bounds__(256) void rownorm_kernel(const float* __restrict__ h,
                                                      float* __restrict__ sqn) {
    const int wid  = threadIdx.x >> 5;
    const int lane = threadIdx.x & 31;
    const int row  = blockIdx.x * 8 + wid;          // 8 waves per block, 1 row per wave
    const float* p = h + (size_t)row * DD;
    float s = 0.f;
    #pragma unroll
    for (int c = 0; c < DD / 128; ++c) {            // 4 x float4 per lane
        float4 v = *(const float4*)(p + lane * 4 + c * 128);
        s += v.x * v.x + v.y * v.y + v.z * v.z + v.w * v.w;
    }
    #pragma unroll
    for (int off = 16; off > 0; off >>= 1)
        s += __shfl_xor(s, off, 32);
    if (lane == 0) sqn[row] = s;
}

// --- Kernel 2: fused  sqI + sqJ - 2 * H H^T  via fp32 WMMA -------------------
// Double-buffered LDS staging through the CDNA5 async global->LDS path
// (GLOBAL_LOAD_ASYNC_TO_LDS_B64, ASYNCcnt), prefetching one K-step ahead.
__global__ __launch_bounds__(256) void pairdist_kernel(const float* __restrict__ h,
                                                       const float* __restrict__ sqn,
                                                       float* __restrict__ out) {
    __shared__ float lA[2][BT * LDSTR];
    __shared__ float lB[2][BT * LDSTR];

    const int g     = blockIdx.z;
    const int iBase = blockIdx.x * BT;
    const int jBase = blockIdx.y * BT;
    const float* hg = h + (size_t)g * NN * DD;

    const int tid   = threadIdx.x;
    const int wid   = tid >> 5;
    const int lane  = tid & 31;
    const int m16   = lane & 15;    // row within 16x16 fragment / output column
    const int kh    = lane >> 4;    // which K-pair half (K=0,1 vs K=2,3)
    const int waveM = wid & 3;      // 4 wave-rows  -> 32-row strips
    const int waveN = wid >> 2;     // 2 wave-cols  -> 64-col strips

    v8f acc[2][4];
    #pragma unroll
    for (int a = 0; a < 2; ++a)
        #pragma unroll
        for (int b = 0; b < 4; ++b)
            #pragma unroll
            for (int e = 0; e < 8; ++e)
                acc[a][b][e] = 0.f;

    // Issue 16 async b64 copies per wave: both 128xBK strips into one LDS buffer.
    auto stage = [&](int k0, float* bufA, float* bufB) {
        #pragma unroll
        for (int it = 0; it < 4; ++it) {
            const int idx = tid + it * 256;         // 0..1023
            const int row = idx >> 3;               // 0..127
            const int q   = idx & 7;                // float4 column within BK
            const float* ga = hg + (size_t)(iBase + row) * DD + k0 + q * 4;
            const float* gb = hg + (size_t)(jBase + row) * DD + k0 + q * 4;
            unsigned la = (unsigned)(uintptr_t)(bufA + row * LDSTR + q * 4);
            unsigned lb = (unsigned)(uintptr_t)(bufB + row * LDSTR + q * 4);
            // IOFFSET is added to BOTH the LDS and global address (ISA 08 §4.4),
            // so one address pair serves both 8-byte halves of the float4.
            asm volatile("global_load_async_to_lds_b64 %0, %1, off"
                         :: "v"(la), "v"(ga) : "memory");
            asm volatile("global_load_async_to_lds_b64 %0, %1, off offset:8"
                         :: "v"(la), "v"(ga) : "memory");
            asm volatile("global_load_async_to_lds_b64 %0, %1, off"
                         :: "v"(lb), "v"(gb) : "memory");
            asm volatile("global_load_async_to_lds_b64 %0, %1, off offset:8"
                         :: "v"(lb), "v"(gb) : "memory");
        }
    };

    auto compute = [&](const float* sA, const float* sB) {
        #pragma unroll
        for (int kk = 0; kk < BK; kk += 4) {
            // ISA 16x4 fp32 A layout: lanes 0-15 hold K=kk,kk+1; lanes 16-31 K=kk+2,kk+3
            v2f af[2], bf[4];
            #pragma unroll
            for (int mt = 0; mt < 2; ++mt) {
                const int row = waveM * 32 + mt * 16 + m16;
                af[mt] = *(const v2f*)&sA[row * LDSTR + kk + 2 * kh];
            }
            #pragma unroll
            for (int nt = 0; nt < 4; ++nt) {
                const int row = waveN * 64 + nt * 16 + m16;
                bf[nt] = *(const v2f*)&sB[row * LDSTR + kk + 2 * kh];
            }
            #pragma unroll
            for (int mt = 0; mt < 2; ++mt)
                #pragma unroll
                for (int nt = 0; nt < 4; ++nt)
                    acc[mt][nt] = __builtin_amdgcn_wmma_f32_16x16x4_f32(
                        false, af[mt], false, bf[nt],
                        (short)0, acc[mt][nt], false, false);
        }
    };

    // Prologue: stage the first K-step into buffer 0.
    stage(0, lA[0], lB[0]);

    // Main loop, 2x unrolled so buffer selection stays compile-time static.
    // Per phase: prefetch next K-step into the other buffer (overwrites the
    // buffer whose compute finished at the previous trailing barrier), wait
    // for the CURRENT buffer's 16 in-order async copies, then compute.
    for (int k0 = 0; k0 < DD; k0 += 2 * BK) {
        // ---- phase A: compute buf0, prefetch buf1 ----
        stage(k0 + BK, lA[1], lB[1]);                      // k0+BK < DD always here
        asm volatile("s_wait_asynccnt 0x10" ::: "memory"); // current 16 done
        __syncthreads();
        compute(lA[0], lB[0]);
        __syncthreads();

        // ---- phase B: compute buf1, prefetch buf0 ----
        if (k0 + 2 * BK < DD) {
            stage(k0 + 2 * BK, lA[0], lB[0]);
            asm volatile("s_wait_asynccnt 0x10" ::: "memory");
        } else {
            asm volatile("s_wait_asynccnt 0x0" ::: "memory");
        }
        __syncthreads();
        compute(lA[1], lB[1]);
        __syncthreads();
    }

    // Epilogue: sm = ||q_i||^2 + ||k_j||^2 - 2 * gram, fused into the store.
    float* og = out + (size_t)g * NN * NN;
    const float* sq = sqn + g * NN;

    float si[2][8];
    #pragma unroll
    for (int mt = 0; mt < 2; ++mt)
        #pragma unroll
        for (int r = 0; r < 8; ++r)
            si[mt][r] = sq[iBase + waveM * 32 + mt * 16 + 8 * kh + r];

    #pragma unroll
    for (int nt = 0; nt < 4; ++nt) {
        const int col = jBase + waveN * 64 + nt * 16 + m16;
        const float sj = sq[col];
        #pragma unroll
        for (int mt = 0; mt < 2; ++mt) {
            const int rbase = iBase + waveM * 32 + mt * 16 + 8 * kh;
            #pragma unroll
            for (int r = 0; r < 8; ++r)
                og[(size_t)(rbase + r) * NN + col] = si[mt][r] + sj - 2.0f * acc[mt][nt][r];
        }
    }
}

extern "C" void kernel_launch(void* const* d_in, const int* in_sizes, int n_in,
                              void* d_out, int out_size, void* d_ws, size_t ws_size,
                              hipStream_t stream) {
    const float* h   = (const float*)d_in[0];   // [G*NN, DD] fp32
    float*       out = (float*)d_out;           // [G, NN, NN] fp32
    float*       sqn = (float*)d_ws;            // [G*NN] fp32 scratch (128 KB)

    // 1) row squared norms: 32768 rows, 8 rows (waves) per 256-thread block
    rownorm_kernel<<<(G * NN) / 8, 256, 0, stream>>>(h, sqn);

    // 2) fused pairwise-distance GEMM: 16x16 tiles of 128x128 per graph
    dim3 grid(NN / BT, NN / BT, G);
    pairdist_kernel<<<grid, 256, 0, stream>>>(h, sqn, out);
}